// PairwiseUnsupervisedLoss_46505905881729
// MI455X (gfx1250) — compile-verified
//
#include <hip/hip_runtime.h>

// ---------------------------------------------------------------------------
// PairwiseUnsupervisedLoss for MI455X (gfx1250, wave32).
// Memory-bound (134 MB traffic, fits 192 MB L2). WMMA f32 16x16x4 is used as a
// one-shot wave32 reduction engine (A = per-lane partials, B = ones).
// ---------------------------------------------------------------------------

typedef float v2f __attribute__((ext_vector_type(2)));
typedef float v8f __attribute__((ext_vector_type(8)));

static constexpr int   B_ = 16, C_ = 3, H_ = 512, W_ = 512;
static constexpr float SSIM_C1 = 0.01f * 0.01f;
static constexpr float SSIM_C2 = 0.03f * 0.03f;
static constexpr float EDGE_ALPHA = 10.0f;

// Full wave32 sum via V_WMMA_F32_16X16X4_F32.
// A layout (16x4 f32): VGPR0 lanes0-15 -> K=0, lanes16-31 -> K=2; VGPR1 -> K=1/K=3.
// With a={v,0}, B=ones: D[m][n] = v_m + v_{m+16}.
// D layout: VGPR j, lanes0-15 -> M=j, lanes16-31 -> M=j+8, so summing the 8 D
// regs gives the half-sum per lane group; one shfl_xor(16) completes the sum.
__device__ __forceinline__ float wave_sum32(float v) {
    v2f a; a[0] = v;    a[1] = 0.0f;
    v2f b; b[0] = 1.0f; b[1] = 1.0f;
    v8f c = {};
    v8f d = __builtin_amdgcn_wmma_f32_16x16x4_f32(
        /*neg_a=*/false, a, /*neg_b=*/false, b,
        /*c_mod=*/(short)0, c, /*reuse_a=*/false, /*reuse_b=*/false);
    float s = d[0] + d[1] + d[2] + d[3] + d[4] + d[5] + d[6] + d[7];
    s += __shfl_xor(s, 16, 32);
    return s;
}

// Block-wide sum of v (every thread must call; EXEC must be all ones for WMMA),
// accumulated into a global double slot.
__device__ __forceinline__ void block_accum(double* target, float v, float* sbuf) {
    const float ws   = wave_sum32(v);
    const int   lane = threadIdx.x & 31;
    const int   wv   = threadIdx.x >> 5;
    if (lane == 0) sbuf[wv] = ws;
    __syncthreads();
    if (threadIdx.x == 0) {
        float t = 0.0f;
        const int nw = blockDim.x >> 5;
        for (int i = 0; i < nw; ++i) t += sbuf[i];
        atomicAdd(target, (double)t);
    }
    __syncthreads();
}

__global__ void k_init(double* __restrict__ acc) {
    if (threadIdx.x < 4) acc[threadIdx.x] = 0.0;
}

// Pass 1: bilinear warp + L1 photometric sum + edge-aware smoothness sums.
__global__ void k_l1_smooth(const float* __restrict__ img1,
                            const float* __restrict__ img2,
                            const float* __restrict__ flow,
                            double* __restrict__ acc) {
    const int HW = H_ * W_;
    const long long N = (long long)B_ * HW;
    float l1 = 0.0f, sx = 0.0f, sy = 0.0f;

    for (long long p = (long long)blockIdx.x * blockDim.x + threadIdx.x; p < N;
         p += (long long)gridDim.x * blockDim.x) {
        const int b = (int)(p / HW);
        const int r = (int)(p % HW);
        const int h = r / W_;
        const int w = r - h * W_;

        const float* f0 = flow + (long long)(b * 2 + 0) * HW;
        const float* f1 = flow + (long long)(b * 2 + 1) * HW;
        const float* i1 = img1 + (long long)b * C_ * HW;
        const float* i2 = img2 + (long long)b * C_ * HW;

        // --- bilinear warp of img2 (matches reference clamping) ---
        const float vx = fminf(fmaxf((float)w + f0[r], 0.0f), (float)(W_ - 1));
        const float vy = fminf(fmaxf((float)h + f1[r], 0.0f), (float)(H_ - 1));
        const float x0f = floorf(vx), y0f = floorf(vy);
        const float wx = vx - x0f, wy = vy - y0f;
        const int x0 = min(max((int)x0f, 0), W_ - 1);
        const int x1 = min(x0 + 1, W_ - 1);
        const int y0 = min(max((int)y0f, 0), H_ - 1);
        const int y1 = min(y0 + 1, H_ - 1);
        const int iA = y0 * W_ + x0, iB = y0 * W_ + x1;
        const int iC = y1 * W_ + x0, iD = y1 * W_ + x1;
        const float w00 = (1.0f - wx) * (1.0f - wy), w01 = wx * (1.0f - wy);
        const float w10 = (1.0f - wx) * wy,          w11 = wx * wy;

#pragma unroll
        for (int c = 0; c < C_; ++c) {
            const float* p2 = i2 + c * HW;
            const float warped = w00 * p2[iA] + w01 * p2[iB] +
                                 w10 * p2[iC] + w11 * p2[iD];
            l1 += fabsf(i1[c * HW + r] - warped);
        }

        // --- smoothness (L2-resident regathers are cheap) ---
        if (w < W_ - 1) {
            const float m = (fabsf(i1[0 * HW + r + 1] - i1[0 * HW + r]) +
                             fabsf(i1[1 * HW + r + 1] - i1[1 * HW + r]) +
                             fabsf(i1[2 * HW + r + 1] - i1[2 * HW + r])) * (1.0f / 3.0f);
            const float wgt = expf(-EDGE_ALPHA * m);
            sx += wgt * (fabsf(f0[r + 1] - f0[r]) + fabsf(f1[r + 1] - f1[r]));
        }
        if (h < H_ - 1) {
            const float m = (fabsf(i1[0 * HW + r + W_] - i1[0 * HW + r]) +
                             fabsf(i1[1 * HW + r + W_] - i1[1 * HW + r]) +
                             fabsf(i1[2 * HW + r + W_] - i1[2 * HW + r])) * (1.0f / 3.0f);
            const float wgt = expf(-EDGE_ALPHA * m);
            sy += wgt * (fabsf(f0[r + W_] - f0[r]) + fabsf(f1[r + W_] - f1[r]));
        }
    }

    __shared__ float sbuf[8];
    block_accum(acc + 0, l1, sbuf);
    block_accum(acc + 2, sx, sbuf);
    block_accum(acc + 3, sy, sbuf);
}

// Pass 2: SSIM over the VALID (H-2)x(W-2) region, 16x16 outputs per block,
// 18x18 halo of img1 and re-warped img2 staged in LDS per channel.
#define TILE 16
#define HALO 18
#define HALO2 (HALO * HALO)

__global__ void k_ssim(const float* __restrict__ img1,
                       const float* __restrict__ img2,
                       const float* __restrict__ flow,
                       double* __restrict__ acc) {
    const int HW = H_ * W_;
    const int HV = H_ - 2, WV = W_ - 2;               // 510 x 510 valid
    const int tilesX = (WV + TILE - 1) / TILE;        // 32
    const int tilesY = (HV + TILE - 1) / TILE;        // 32

    const int b  = blockIdx.x / (tilesX * tilesY);
    const int t  = blockIdx.x % (tilesX * tilesY);
    const int oy = (t / tilesX) * TILE;
    const int ox = (t % tilesX) * TILE;

    __shared__ int   sPix[HALO2];
    __shared__ int   sIA[HALO2], sIB[HALO2], sIC[HALO2], sID[HALO2];
    __shared__ float sW00[HALO2], sW01[HALO2], sW10[HALO2], sW11[HALO2];
    __shared__ float sX[HALO2], sY[HALO2];
    __shared__ float sbuf[8];

    const float* f0 = flow + (long long)(b * 2 + 0) * HW;
    const float* f1 = flow + (long long)(b * 2 + 1) * HW;

    // Precompute bilinear gather offsets/weights once per halo pixel.
    for (int idx = threadIdx.x; idx < HALO2; idx += blockDim.x) {
        int iy = oy + idx / HALO; if (iy > H_ - 1) iy = H_ - 1;
        int ix = ox + idx % HALO; if (ix > W_ - 1) ix = W_ - 1;
        const int r = iy * W_ + ix;
        sPix[idx] = r;
        const float vx = fminf(fmaxf((float)ix + f0[r], 0.0f), (float)(W_ - 1));
        const float vy = fminf(fmaxf((float)iy + f1[r], 0.0f), (float)(H_ - 1));
        const float x0f = floorf(vx), y0f = floorf(vy);
        const float wx = vx - x0f, wy = vy - y0f;
        const int x0 = min(max((int)x0f, 0), W_ - 1);
        const int x1 = min(x0 + 1, W_ - 1);
        const int y0 = min(max((int)y0f, 0), H_ - 1);
        const int y1 = min(y0 + 1, H_ - 1);
        sIA[idx] = y0 * W_ + x0;  sIB[idx] = y0 * W_ + x1;
        sIC[idx] = y1 * W_ + x0;  sID[idx] = y1 * W_ + x1;
        sW00[idx] = (1.0f - wx) * (1.0f - wy);  sW01[idx] = wx * (1.0f - wy);
        sW10[idx] = (1.0f - wx) * wy;           sW11[idx] = wx * wy;
    }
    __syncthreads();

    const int  ly    = threadIdx.x / TILE;
    const int  lx    = threadIdx.x % TILE;
    const bool valid = (oy + ly < HV) && (ox + lx < WV);
    float ssim_acc = 0.0f;

    for (int c = 0; c < C_; ++c) {
        const float* i1c = img1 + ((long long)b * C_ + c) * HW;
        const float* i2c = img2 + ((long long)b * C_ + c) * HW;
        for (int idx = threadIdx.x; idx < HALO2; idx += blockDim.x) {
            sX[idx] = i1c[sPix[idx]];
            sY[idx] = sW00[idx] * i2c[sIA[idx]] + sW01[idx] * i2c[sIB[idx]] +
                      sW10[idx] * i2c[sIC[idx]] + sW11[idx] * i2c[sID[idx]];
        }
        __syncthreads();

        float ax = 0.f, ay = 0.f, axx = 0.f, ayy = 0.f, axy = 0.f;
#pragma unroll
        for (int di = 0; di < 3; ++di) {
#pragma unroll
            for (int dj = 0; dj < 3; ++dj) {
                const float x = sX[(ly + di) * HALO + (lx + dj)];
                const float y = sY[(ly + di) * HALO + (lx + dj)];
                ax += x; ay += y; axx += x * x; ayy += y * y; axy += x * y;
            }
        }
        const float inv9 = 1.0f / 9.0f;
        const float mux = ax * inv9, muy = ay * inv9;
        const float mux2 = mux * mux, muy2 = muy * muy, muxy = mux * muy;
        const float sgx  = axx * inv9 - mux2;
        const float sgy  = ayy * inv9 - muy2;
        const float sgxy = axy * inv9 - muxy;
        const float n = (2.0f * muxy + SSIM_C1) * (2.0f * sgxy + SSIM_C2);
        const float d = (mux2 + muy2 + SSIM_C1) * (sgx + sgy + SSIM_C2);
        float v = 0.5f * (1.0f - n / d);
        v = fminf(fmaxf(v, 0.0f), 1.0f);
        if (valid) ssim_acc += v;
        __syncthreads();   // before LDS reuse next channel
    }

    block_accum(acc + 1, ssim_acc, sbuf);
}

__global__ void k_final(const double* __restrict__ acc, float* __restrict__ out) {
    if (threadIdx.x == 0 && blockIdx.x == 0) {
        const double l1   = acc[0] / ((double)B_ * C_ * H_ * W_);
        const double ssim = acc[1] / ((double)B_ * C_ * (H_ - 2) * (W_ - 2));
        const double photo = 0.15 * l1 + 0.85 * ssim;
        // loss_x elem = wgt*|dx|/2; our acc[2] = sum(wgt*(|dx0|+|dx1|)) = 2*sum(loss_x)
        // smooth = mean(loss_x)/2 + mean(loss_y)/2
        const double sx = acc[2] / (4.0 * (double)B_ * 2.0 * H_ * (W_ - 1));
        const double sy = acc[3] / (4.0 * (double)B_ * 2.0 * (H_ - 1) * W_);
        const double smooth = sx + sy;
        const double total  = photo + 10.0 * smooth;
        out[0] = (float)total;
        out[1] = (float)photo;
        out[2] = (float)smooth;
    }
}

extern "C" void kernel_launch(void* const* d_in, const int* in_sizes, int n_in,
                              void* d_out, int out_size, void* d_ws, size_t ws_size,
                              hipStream_t stream) {
    (void)in_sizes; (void)n_in; (void)out_size; (void)ws_size;
    const float* img1 = (const float*)d_in[0];
    const float* img2 = (const float*)d_in[1];
    const float* flow = (const float*)d_in[2];
    double* acc = (double*)d_ws;          // [l1, ssim, smooth_x, smooth_y]
    float*  out = (float*)d_out;

    k_init<<<1, 32, 0, stream>>>(acc);
    k_l1_smooth<<<4096, 256, 0, stream>>>(img1, img2, flow, acc);
    const int tiles = ((W_ - 2 + TILE - 1) / TILE) * ((H_ - 2 + TILE - 1) / TILE); // 1024
    k_ssim<<<B_ * tiles, 256, 0, stream>>>(img1, img2, flow, acc);
    k_final<<<1, 32, 0, stream>>>(acc, out);
}